// Attention_88012469830325
// MI455X (gfx1250) — compile-verified
//
#include <hip/hip_runtime.h>
#include <hip/hip_bf16.h>

typedef __attribute__((ext_vector_type(2))) float v2f;
typedef __attribute__((ext_vector_type(4))) float f4;
typedef __attribute__((ext_vector_type(8))) float v8f;

#define WMMA_F32(a, b, c) \
    __builtin_amdgcn_wmma_f32_16x16x4_f32(false, (a), false, (b), (short)0, (c), false, false)

// ---------------------------------------------------------------------------
// GEMM: C[m,n] = sum_k A[m,k] * Bt[n,k]  (+ bias[n] if bias != null)
// A: M x K row-major, Bt: N x K row-major (B transposed), C: M x N.
// One wave computes a 32x32 patch of C = 2x2 tiles of 16x16 via
// V_WMMA_F32_16X16X4_F32, reusing each A/B fragment twice
// (1 b64 load per WMMA -> 8 FLOP/byte vs L2).
// Explicit register double-buffering: prefetch next 16-wide K chunk
// (16 b64 loads) before computing the current chunk's 16 WMMAs, so WMMAs
// only wait on loads issued a full chunk earlier.
// A-frag layout (ISA 7.12.2, 32-bit A 16x4): lanes 0-15 hold K={0,1},
// lanes 16-31 hold K={2,3} -> per lane one contiguous 2-float (b64) load.
// B-frag mirrors it over N. C/D layout: VGPR i, lanes 0-15 -> M=i,
// lanes 16-31 -> M=8+i, N = lane&15.
// ---------------------------------------------------------------------------
__global__ void wmma_gemm_f32(const float* __restrict__ A,
                              const float* __restrict__ Bt,
                              const float* __restrict__ bias,
                              float* __restrict__ C,
                              int M, int N, int K) {
    const int lane = threadIdx.x & 31;
    const int waveInBlk = threadIdx.x >> 5;
    const int wavesPerBlk = blockDim.x >> 5;
    const int wave = blockIdx.x * wavesPerBlk + waveInBlk;

    const int patchesN = N >> 5;                 // 32-wide patches
    const int m0 = (wave / patchesN) << 5;
    const int n0 = (wave % patchesN) << 5;
    if (m0 >= M) return;

    const int l16 = lane & 15;
    const int hi  = lane >> 4;          // 0: K lanes {0,1}; 1: K lanes {2,3}

    const float* aPtr0 = A  + (size_t)(m0 + l16) * K + 2 * hi;
    const float* aPtr1 = A  + (size_t)(m0 + 16 + l16) * K + 2 * hi;
    const float* bPtr0 = Bt + (size_t)(n0 + l16) * K + 2 * hi;
    const float* bPtr1 = Bt + (size_t)(n0 + 16 + l16) * K + 2 * hi;

    v8f acc00 = {}, acc01 = {}, acc10 = {}, acc11 = {};

    // current-chunk fragments (held in registers; indices are unroll-constant)
    v2f ca0[4], ca1[4], cb0[4], cb1[4];
#pragma unroll
    for (int u = 0; u < 4; ++u) {
        ca0[u] = *(const v2f*)(aPtr0 + 4 * u);
        ca1[u] = *(const v2f*)(aPtr1 + 4 * u);
        cb0[u] = *(const v2f*)(bPtr0 + 4 * u);
        cb1[u] = *(const v2f*)(bPtr1 + 4 * u);
    }

    int k = 0;
    for (; k + 16 < K; k += 16) {
        // prefetch next chunk (issues 16 b64 loads before any wmma below)
        v2f na0[4], na1[4], nb0[4], nb1[4];
#pragma unroll
        for (int u = 0; u < 4; ++u) {
            const int kk = k + 16 + 4 * u;
            na0[u] = *(const v2f*)(aPtr0 + kk);
            na1[u] = *(const v2f*)(aPtr1 + kk);
            nb0[u] = *(const v2f*)(bPtr0 + kk);
            nb1[u] = *(const v2f*)(bPtr1 + kk);
        }
        // compute on current chunk
#pragma unroll
        for (int u = 0; u < 4; ++u) {
            acc00 = WMMA_F32(ca0[u], cb0[u], acc00);
            acc01 = WMMA_F32(ca0[u], cb1[u], acc01);
            acc10 = WMMA_F32(ca1[u], cb0[u], acc10);
            acc11 = WMMA_F32(ca1[u], cb1[u], acc11);
        }
        // rotate buffers
#pragma unroll
        for (int u = 0; u < 4; ++u) {
            ca0[u] = na0[u]; ca1[u] = na1[u];
            cb0[u] = nb0[u]; cb1[u] = nb1[u];
        }
    }
    // epilogue: last chunk
#pragma unroll
    for (int u = 0; u < 4; ++u) {
        acc00 = WMMA_F32(ca0[u], cb0[u], acc00);
        acc01 = WMMA_F32(ca0[u], cb1[u], acc01);
        acc10 = WMMA_F32(ca1[u], cb0[u], acc10);
        acc11 = WMMA_F32(ca1[u], cb1[u], acc11);
    }

    const float bv0 = bias ? bias[n0 + l16] : 0.0f;
    const float bv1 = bias ? bias[n0 + 16 + l16] : 0.0f;

    float* c00 = C + (size_t)(m0 + 8 * hi) * N + n0 + l16;
    float* c01 = c00 + 16;
    float* c10 = C + (size_t)(m0 + 16 + 8 * hi) * N + n0 + l16;
    float* c11 = c10 + 16;
#pragma unroll
    for (int i = 0; i < 8; ++i) {
        c00[(size_t)i * N] = acc00[i] + bv0;
        c01[(size_t)i * N] = acc01[i] + bv1;
        c10[(size_t)i * N] = acc10[i] + bv0;
        c11[(size_t)i * N] = acc11[i] + bv1;
    }
}

// ---------------------------------------------------------------------------
// Per-head BatchNorm statistics over v (= qv[:, 2048:]):
// mean/var over 512*128 elements per head; fold gamma/beta into scale/shift.
// v element (b,h,d) lives at vBase[b*4096 + h*128 + d].
// ---------------------------------------------------------------------------
__global__ void bn_stats(const float* __restrict__ vBase,
                         const float* __restrict__ gamma,
                         const float* __restrict__ beta,
                         float* __restrict__ scale,
                         float* __restrict__ shift) {
    const int h = blockIdx.x;        // 16 heads
    const int t = threadIdx.x;       // 256 threads
    float s = 0.0f, s2 = 0.0f;
    for (int i = t; i < 512 * 128; i += 256) {
        const int b = i >> 7;
        const int d = i & 127;
        const float x = vBase[(size_t)b * 4096 + h * 128 + d];
        s  += x;
        s2 += x * x;
    }
    __shared__ float ss[256];
    __shared__ float ss2[256];
    ss[t] = s; ss2[t] = s2;
    __syncthreads();
    for (int off = 128; off > 0; off >>= 1) {
        if (t < off) { ss[t] += ss[t + off]; ss2[t] += ss2[t + off]; }
        __syncthreads();
    }
    if (t == 0) {
        const float inv_n = 1.0f / (512.0f * 128.0f);
        const float mean = ss[0] * inv_n;
        const float var  = ss2[0] * inv_n - mean * mean;
        const float rstd = rsqrtf(var + 1e-5f);
        const float sc = gamma[h] * rstd;
        scale[h] = sc;
        shift[h] = beta[h] - mean * sc;
    }
}

// ---------------------------------------------------------------------------
// Attention: for each (b,h,c):
//   s_d  = temperature[h] * (q[c]*k[d] + bias[b,h,c,d])
//   out[c] = sum_d softmax(s)_d * v_norm[d]
// One block per (b,h). k, v_norm, q staged in LDS. Each wave handles rows c
// (8 waves x 16 rows); each lane covers 4 of the 128 d values (b128 bias
// load, nontemporal: bias is 536MB streamed once, bigger than L2).
// Wave32 butterfly reductions for max / sum / dot. attn matrix is never
// materialized in memory (saves ~1GB of round-trip traffic).
// ---------------------------------------------------------------------------
__global__ void attn_softmax(const float* __restrict__ qv,      // 512 x 4096
                             const float* __restrict__ k_param, // 512 x 2048
                             const float* __restrict__ bias,    // 512*16*128*128
                             const float* __restrict__ temp,    // 16
                             const float* __restrict__ scale,   // 16
                             const float* __restrict__ shift,   // 16
                             float* __restrict__ out) {         // 512 x 2048
    const int bh = blockIdx.x;       // 0..8191
    const int b  = bh >> 4;
    const int h  = bh & 15;
    const int t  = threadIdx.x;      // 256 = 8 waves
    const int lane = t & 31;
    const int wave = t >> 5;

    __shared__ float k_s[128];
    __shared__ float v_s[128];
    __shared__ float q_s[128];

    const float sc = scale[h];
    const float sh = shift[h];
    const float tp = temp[h];

    if (t < 128) {
        k_s[t] = k_param[(size_t)b * 2048 + h * 128 + t];
        const float vv = qv[(size_t)b * 4096 + 2048 + h * 128 + t];
        v_s[t] = vv * sc + sh;
        q_s[t] = qv[(size_t)b * 4096 + h * 128 + t];
    }
    __syncthreads();

    const float* biasBH = bias + (size_t)bh * (128 * 128);
    const int d0 = lane * 4;
    const float k0 = k_s[d0 + 0], k1 = k_s[d0 + 1];
    const float k2 = k_s[d0 + 2], k3 = k_s[d0 + 3];
    const float w0 = v_s[d0 + 0], w1 = v_s[d0 + 1];
    const float w2 = v_s[d0 + 2], w3 = v_s[d0 + 3];

    for (int c = wave; c < 128; c += 8) {
        const float qc = q_s[c] * tp;
        const f4 bv = __builtin_nontemporal_load(
            (const f4*)(biasBH + (size_t)c * 128) + lane);

        float s0 = qc * k0 + tp * bv.x;
        float s1 = qc * k1 + tp * bv.y;
        float s2 = qc * k2 + tp * bv.z;
        float s3 = qc * k3 + tp * bv.w;

        float m = fmaxf(fmaxf(s0, s1), fmaxf(s2, s3));
#pragma unroll
        for (int off = 16; off > 0; off >>= 1)
            m = fmaxf(m, __shfl_xor(m, off, 32));

        const float e0 = __expf(s0 - m);
        const float e1 = __expf(s1 - m);
        const float e2 = __expf(s2 - m);
        const float e3 = __expf(s3 - m);

        float sum = e0 + e1 + e2 + e3;
        float acc = e0 * w0 + e1 * w1 + e2 * w2 + e3 * w3;
#pragma unroll
        for (int off = 16; off > 0; off >>= 1) {
            sum += __shfl_xor(sum, off, 32);
            acc += __shfl_xor(acc, off, 32);
        }

        if (lane == 0)
            out[(size_t)b * 2048 + h * 128 + c] = acc / sum;
    }
}

// ---------------------------------------------------------------------------
// Launch: qv-GEMM -> BN stats -> attention/softmax -> out-GEMM (+bias)
// ---------------------------------------------------------------------------
extern "C" void kernel_launch(void* const* d_in, const int* in_sizes, int n_in,
                              void* d_out, int out_size, void* d_ws, size_t ws_size,
                              hipStream_t stream) {
    const float* x          = (const float*)d_in[0]; // 512*2048
    const float* W_qv       = (const float*)d_in[1]; // 4096*2048
    const float* temperature= (const float*)d_in[2]; // 16
    const float* k_param    = (const float*)d_in[3]; // 512*2048
    const float* bias_param = (const float*)d_in[4]; // 512*16*128*128
    const float* bn_gamma   = (const float*)d_in[5]; // 16
    const float* bn_beta    = (const float*)d_in[6]; // 16
    const float* W_out      = (const float*)d_in[7]; // 2048*2048
    const float* b_out      = (const float*)d_in[8]; // 2048
    float* out = (float*)d_out;                      // 512*2048

    float* qv       = (float*)d_ws;            // 512*4096 f32 = 8 MB
    float* attn_out = qv + 512 * 4096;         // 512*2048 f32 = 4 MB
    float* scale    = attn_out + 512 * 2048;   // 16 f32
    float* shift    = scale + 16;              // 16 f32

    // GEMM1: qv = x @ W_qv^T   (M=512, N=4096, K=2048)
    {
        const int waves = (512 / 32) * (4096 / 32);   // 2048 waves (32x32 each)
        wmma_gemm_f32<<<waves / 8, 256, 0, stream>>>(
            x, W_qv, nullptr, qv, 512, 4096, 2048);
    }

    // Per-head BN stats over v = qv[:, 2048:]
    bn_stats<<<16, 256, 0, stream>>>(qv + 2048, bn_gamma, bn_beta, scale, shift);

    // Attention + softmax + attn@v  (streams the 536MB bias tensor)
    attn_softmax<<<512 * 16, 256, 0, stream>>>(
        qv, k_param, bias_param, temperature, scale, shift, attn_out);

    // GEMM2: out = attn_out @ W_out^T + b_out  (M=512, N=2048, K=2048)
    {
        const int waves = (512 / 32) * (2048 / 32);   // 1024 waves
        wmma_gemm_f32<<<waves / 8, 256, 0, stream>>>(
            attn_out, W_out, b_out, out, 512, 2048, 2048);
    }
}